// ModulatedConv2D_2327872274501
// MI455X (gfx1250) — compile-verified
//
#include <hip/hip_runtime.h>
#include <hip/hip_bf16.h>

// ---------------- problem constants ----------------
#define BATCH   16
#define INCH    512
#define OUTCH   512
#define HH      64
#define WW      64
#define KKSQ    9           // 3*3
#define KTOT    (INCH*KKSQ) // 4608

#define LIN_SCALE  0.04419417382415922f   // sqrt(2)/sqrt(1024) = 1/sqrt(512)
#define CONV_SCALE 0.020833333333333332f  // sqrt(2)/sqrt(4608) = 1/48

typedef float v2f __attribute__((ext_vector_type(2)));
typedef float v8f __attribute__((ext_vector_type(8)));

// ---------------- kernel 1: style linear  s[b,i] = lin_b[i] + LIN_SCALE * <y[b,:], lin_w[i,:]> ----
__global__ void style_kernel(const float* __restrict__ y,
                             const float* __restrict__ lin_w,
                             const float* __restrict__ lin_b,
                             float* __restrict__ s) {
    int t = blockIdx.x * blockDim.x + threadIdx.x;   // 8192 threads
    int b = t & (BATCH - 1);
    int i = t >> 4;
    const float* yp = y + b * INCH;
    const float* wp = lin_w + i * INCH;              // row i contiguous, broadcast across 16 lanes
    float sum = 0.f;
#pragma unroll 8
    for (int j = 0; j < INCH; ++j) sum += yp[j] * wp[j];
    s[b * INCH + i] = lin_b[i] + LIN_SCALE * sum;
}

// ---------------- kernel 2: demod norm[b,o] = rsqrt(CONV_SCALE^2 * sum_i (sum_kk w^2) * s^2 + eps) ----
__global__ void demod_kernel(const float* __restrict__ conv_w,
                             const float* __restrict__ s,
                             float* __restrict__ norm) {
    int t = blockIdx.x * blockDim.x + threadIdx.x;   // 8192 threads
    int b = t & (BATCH - 1);
    int o = t >> 4;                                  // 16 lanes share one weight row -> L2 broadcast
    const float* wrow = conv_w + (size_t)o * KTOT;   // contiguous 4608 floats
    const float* sp   = s + b * INCH;
    float sum = 0.f;
    for (int i = 0; i < INCH; ++i) {
        float sv = sp[i];
        const float* wp = wrow + i * KKSQ;
        float wsq = 0.f;
#pragma unroll
        for (int k = 0; k < KKSQ; ++k) wsq += wp[k] * wp[k];
        sum += wsq * (sv * sv);
    }
    norm[b * OUTCH + o] = rsqrtf(sum * (CONV_SCALE * CONV_SCALE) + 1e-8f);
}

// ---------------- kernel 3: modulated conv, f32 WMMA implicit GEMM, double-buffered ----------------
// Block tile: 64 outch x 128 spatial (2 image rows x 64 cols), one batch element.
// 256 threads = 8 waves; wave = (wo2, wn): wo2 = wave>>2 selects 32-outch half (2 subtiles),
// wn = wave&3 selects (row = wn>>1, colbase = (wn&1)*32) -> 2 column subtiles.
// Each wave: 2x2 grid of 16x16 WMMA tiles -> 4 WMMAs share 2 A + 2 B fragments
// (2 LDS floats per lane per WMMA). K streamed in 16-channel chunks, K-order kk*16+ic,
// double-buffered LDS so chunk c+1 staging overlaps chunk c WMMAs (1 barrier/iter).

#define IC_TILE    16
#define WK         (KKSQ * IC_TILE)    // 144 K values per chunk
#define W_STRIDE   148                 // padded: L*148 mod 64 distinct for 16 lanes
#define X_COLS     66                  // 64 + 2 halo
#define X_CSTRIDE  68                  // padded row stride
#define X_ROWS     4                   // 2 output rows + 2 halo
#define X_ICSTRIDE (X_ROWS * X_CSTRIDE) // 272 floats per input channel
#define X_ELEMS    (IC_TILE * X_ROWS * X_COLS)  // 4224 staged elements
#define W_ELEMS    (64 * WK)                    // 9216 staged elements

__global__ __launch_bounds__(256)
void modconv_kernel(const float* __restrict__ x,
                    const float* __restrict__ conv_w,
                    const float* __restrict__ s,
                    const float* __restrict__ norm,
                    float* __restrict__ out) {
    __shared__ float Wlds[2][64 * W_STRIDE];          // 2 x 37.9 KB
    __shared__ float Xlds[2][IC_TILE * X_ICSTRIDE];   // 2 x 17.4 KB   (total 110.6 KB)

    const int h0    = blockIdx.x * 2;     // first of 2 image rows
    const int oBase = blockIdx.y * 64;    // outch tile
    const int b     = blockIdx.z;         // batch

    const int tid  = threadIdx.x;
    const int lane = tid & 31;
    const int wave = tid >> 5;
    const int wo2  = wave >> 2;           // 0..1 : 32-outch half
    const int wn   = wave & 3;            // 0..3 : spatial quarter
    const int rloc = wn >> 1;             // local output row 0..1
    const int cb   = (wn & 1) * 32;       // column base 0 or 32
    const int m    = lane & 15;
    const int half = lane >> 4;           // lane half selects K sub-pair / C-row offset

    const float* sb = s + b * INCH;

    v8f acc00 = {}, acc01 = {}, acc10 = {}, acc11 = {};

    // ---- staging helper (as macro-ish lambda): chunk icBase -> buffer buf ----
    auto stage = [&](int icBase, int buf) {
        for (int idx = tid; idx < X_ELEMS; idx += 256) {
            int icl = idx / (X_ROWS * X_COLS);
            int rem = idx - icl * (X_ROWS * X_COLS);
            int r   = rem / X_COLS;
            int c   = rem - r * X_COLS;
            int gh  = h0 + r - 1;
            int gw  = c - 1;
            float v = 0.f;
            if ((unsigned)gh < HH && (unsigned)gw < WW) {
                int icg = icBase + icl;
                v = x[(((size_t)b * INCH + icg) * HH + gh) * WW + gw] * sb[icg];
            }
            Xlds[buf][icl * X_ICSTRIDE + r * X_CSTRIDE + c] = v;
        }
        for (int idx = tid; idx < W_ELEMS; idx += 256) {
            int ol = idx / WK;
            int k  = idx - ol * WK;       // k = kk*16 + icl
            int kk  = k >> 4;
            int icl = k & 15;
            float v = conv_w[(((size_t)(oBase + ol) * INCH) + icBase + icl) * KKSQ + kk];
            Wlds[buf][ol * W_STRIDE + k] = v * CONV_SCALE;
        }
    };

    stage(0, 0);
    __syncthreads();

    const int aRow0 = wo2 * 32 + m;       // A subtile 0 row; subtile 1 at +16
    for (int chunk = 0; chunk < INCH / IC_TILE; ++chunk) {
        const int cur = chunk & 1;
        if (chunk + 1 < INCH / IC_TILE)   // overlap next-chunk staging with WMMAs
            stage((chunk + 1) * IC_TILE, cur ^ 1);

        const float* Wl = Wlds[cur];
        const float* Xl = Xlds[cur];
#pragma unroll
        for (int kk = 0; kk < KKSQ; ++kk) {
            const int kh = kk / 3, kw = kk - 3 * kh;
            const float* xrow = Xl + (rloc + kh) * X_CSTRIDE + kw;
#pragma unroll
            for (int q = 0; q < 4; ++q) {
                const int k0 = kk * 16 + q * 4;
                // A (16x4 f32): lanes 0-15 -> K=k0,k0+1 ; lanes 16-31 -> K=k0+2,k0+3
                v2f a0, a1;
                a0.x = Wl[(aRow0     ) * W_STRIDE + k0 + half * 2 + 0];
                a0.y = Wl[(aRow0     ) * W_STRIDE + k0 + half * 2 + 1];
                a1.x = Wl[(aRow0 + 16) * W_STRIDE + k0 + half * 2 + 0];
                a1.y = Wl[(aRow0 + 16) * W_STRIDE + k0 + half * 2 + 1];
                // B (4x16 f32): ic pair selected by lane half
                const int icl = q * 4 + half * 2;
                const float* xp0 = xrow + (icl + 0) * X_ICSTRIDE;
                const float* xp1 = xrow + (icl + 1) * X_ICSTRIDE;
                v2f b0, b1;
                b0.x = xp0[cb + m];       b0.y = xp1[cb + m];
                b1.x = xp0[cb + 16 + m];  b1.y = xp1[cb + 16 + m];

                acc00 = __builtin_amdgcn_wmma_f32_16x16x4_f32(false, a0, false, b0, (short)0, acc00, false, false);
                acc01 = __builtin_amdgcn_wmma_f32_16x16x4_f32(false, a0, false, b1, (short)0, acc01, false, false);
                acc10 = __builtin_amdgcn_wmma_f32_16x16x4_f32(false, a1, false, b0, (short)0, acc10, false, false);
                acc11 = __builtin_amdgcn_wmma_f32_16x16x4_f32(false, a1, false, b1, (short)0, acc11, false, false);
            }
        }
        __syncthreads();   // staging of chunk+1 done everywhere; buf cur free for chunk+2
    }

    // ---- epilogue: apply demod norm, NT-store (don't evict L2-resident x) ----
    const int h = h0 + rloc;
#pragma unroll
    for (int r = 0; r < 8; ++r) {
        int o0 = oBase + wo2 * 32 + r + half * 8;    // C/D layout: M = vgpr + 8*laneHalf
        int o1 = o0 + 16;
        float n0 = norm[b * OUTCH + o0];
        float n1 = norm[b * OUTCH + o1];
        size_t base0 = (((size_t)b * OUTCH + o0) * HH + h) * WW;
        size_t base1 = (((size_t)b * OUTCH + o1) * HH + h) * WW;
        __builtin_nontemporal_store(acc00[r] * n0, &out[base0 + cb + m]);
        __builtin_nontemporal_store(acc01[r] * n0, &out[base0 + cb + 16 + m]);
        __builtin_nontemporal_store(acc10[r] * n1, &out[base1 + cb + m]);
        __builtin_nontemporal_store(acc11[r] * n1, &out[base1 + cb + 16 + m]);
    }
}

// ---------------- launcher ----------------
extern "C" void kernel_launch(void* const* d_in, const int* in_sizes, int n_in,
                              void* d_out, int out_size, void* d_ws, size_t ws_size,
                              hipStream_t stream) {
    const float* x      = (const float*)d_in[0];
    const float* y      = (const float*)d_in[1];
    const float* conv_w = (const float*)d_in[2];   // (1,512,512,3,3) flat
    const float* lin_w  = (const float*)d_in[3];
    const float* lin_b  = (const float*)d_in[4];
    float* out = (float*)d_out;

    // workspace: s (16*512 f32) then norm (16*512 f32) = 64 KB
    float* s_ws    = (float*)d_ws;
    float* norm_ws = s_ws + BATCH * INCH;

    style_kernel<<<dim3((BATCH * INCH) / 256), dim3(256), 0, stream>>>(y, lin_w, lin_b, s_ws);
    demod_kernel<<<dim3((BATCH * OUTCH) / 256), dim3(256), 0, stream>>>(conv_w, s_ws, norm_ws);

    dim3 grid(HH / 2, OUTCH / 64, BATCH);   // (32, 8, 16) = 4096 blocks
    modconv_kernel<<<grid, dim3(256), 0, stream>>>(x, conv_w, s_ws, norm_ws, out);
}